// GCNConv_81020263072096
// MI455X (gfx1250) — compile-verified
//
#include <hip/hip_runtime.h>
#include <hip/hip_bf16.h>

typedef __attribute__((ext_vector_type(2))) float v2f;
typedef __attribute__((ext_vector_type(8))) float v8f;

#define DDIM 128

// ---------------------------------------------------------------- degrees ---
__global__ __launch_bounds__(256) void deg_init_kernel(int* __restrict__ deg, int n) {
    int i = blockIdx.x * 256 + threadIdx.x;
    if (i < n) deg[i] = 1;  // self loop contributes 1 to every node's degree
}

__global__ __launch_bounds__(256) void deg_accum_kernel(const long long* __restrict__ rows,
                                                        const long long* __restrict__ cols,
                                                        int* __restrict__ deg, int n_edges) {
    int e = blockIdx.x * 256 + threadIdx.x;
    if (e >= n_edges) return;
    long long r = rows[e];
    long long c = cols[e];
    if (r != c) atomicAdd(&deg[(int)r], 1);  // hw global_atomic_add_u32, exact
}

__global__ __launch_bounds__(256) void dinv_kernel(const int* __restrict__ deg,
                                                   float* __restrict__ dinv, int n) {
    int i = blockIdx.x * 256 + threadIdx.x;
    if (i >= n) return;
    int d = deg[i];
    dinv[i] = (d > 0) ? rsqrtf((float)d) : 0.0f;
}

// ------------------------------------------------------------------- GEMM ---
// xw[N,128] = x[N,128] @ W[128,128] using V_WMMA_F32_16X16X4_F32.
// Block: 256 threads = 8 waves; block covers 128 rows, each wave a 16x128 strip.
// W staged in LDS as float2 pairs over K: ldsW[(K/2)*128 + n] = {W[K][n], W[K+1][n]}
// so a B fragment (lanes 0-15: K=k,k+1 / lanes 16-31: K=k+2,k+3 at column n)
// is a single 8-byte LDS read per lane.
__global__ __launch_bounds__(256) void gemm_xw_kernel(const float* __restrict__ x,
                                                      const float* __restrict__ w,
                                                      float* __restrict__ xw, int n_nodes) {
    __shared__ v2f ldsW[64 * DDIM];  // 64 KB

    // Cooperative load of W into LDS, packed as (K even, K odd) pairs per column.
    for (int i = threadIdx.x; i < 64 * DDIM; i += 256) {
        int kk2 = i >> 7;      // K/2
        int n   = i & 127;
        v2f t;
        t.x = w[(2 * kk2 + 0) * DDIM + n];
        t.y = w[(2 * kk2 + 1) * DDIM + n];
        ldsW[i] = t;
    }
    __syncthreads();

    const int wave = threadIdx.x >> 5;
    const int lane = threadIdx.x & 31;
    const int hi   = lane >> 4;   // 0: lanes 0-15, 1: lanes 16-31
    const int lp   = lane & 15;

    const int m0   = blockIdx.x * 128 + wave * 16;
    // Clamp instead of predicating: WMMA requires EXEC all-ones.
    int rowA = m0 + lp;
    int rowC = rowA < (n_nodes - 1) ? rowA : (n_nodes - 1);
    const float* xrow = x + (size_t)rowC * DDIM;

    v8f acc[8];
#pragma unroll
    for (int nt = 0; nt < 8; ++nt) acc[nt] = (v8f){0.f, 0.f, 0.f, 0.f, 0.f, 0.f, 0.f, 0.f};

    for (int k = 0; k < DDIM; k += 4) {
        // A fragment (16x4): lanes 0-15 hold {K=k,k+1}, lanes 16-31 {K=k+2,k+3}.
        v2f a = *reinterpret_cast<const v2f*>(xrow + k + 2 * hi);
        const int bbase = ((k >> 1) + hi) * DDIM + lp;
#pragma unroll
        for (int nt = 0; nt < 8; ++nt) {
            v2f b = ldsW[bbase + nt * 16];
            acc[nt] = __builtin_amdgcn_wmma_f32_16x16x4_f32(
                false, a, false, b, (short)0, acc[nt], false, false);
        }
    }

    // D layout: lanes 0-15 -> M=vgpr, N=lane; lanes 16-31 -> M=vgpr+8, N=lane-16.
#pragma unroll
    for (int nt = 0; nt < 8; ++nt) {
        int n = nt * 16 + lp;
#pragma unroll
        for (int v = 0; v < 8; ++v) {
            int r = m0 + v + 8 * hi;
            if (r < n_nodes) xw[(size_t)r * DDIM + n] = acc[nt][v];
        }
    }
}

// ---------------------------------------------------- init out + self loop ---
__global__ __launch_bounds__(256) void out_init_kernel(const float* __restrict__ xw,
                                                       const float* __restrict__ dinv,
                                                       const float* __restrict__ bias,
                                                       float* __restrict__ out, int n_nodes) {
    int t = blockIdx.x * 256 + threadIdx.x;
    int i = t >> 5;        // node
    int q = t & 31;        // float4 chunk within the 128-wide row
    if (i >= n_nodes) return;
    float s = dinv[i] * dinv[i];  // self-loop norm: dinv[i]*1*dinv[i]
    float4 v = reinterpret_cast<const float4*>(xw + (size_t)i * DDIM)[q];
    float4 b = reinterpret_cast<const float4*>(bias)[q];
    float4 o;
    o.x = b.x + s * v.x;
    o.y = b.y + s * v.y;
    o.z = b.z + s * v.z;
    o.w = b.w + s * v.w;
    reinterpret_cast<float4*>(out + (size_t)i * DDIM)[q] = o;
}

// ------------------------------------------------------------------- SpMM ---
// One wave (32 lanes) per edge; each lane owns a float4 chunk of the 128-wide
// feature row. xw (51 MB) is L2-resident on MI455X (192 MB L2) -> gathers and
// the fp32 hw atomics are served from L2.
__global__ __launch_bounds__(256) void spmm_edge_kernel(const long long* __restrict__ rows,
                                                        const long long* __restrict__ cols,
                                                        const float* __restrict__ dinv,
                                                        const float* __restrict__ xw,
                                                        float* __restrict__ out, int n_edges) {
    int e = blockIdx.x * 8 + (threadIdx.x >> 5);
    if (e >= n_edges) return;
    int lane = threadIdx.x & 31;
    if (lane == 0) {
        __builtin_prefetch(rows + e + 64, 0, 0);  // global_prefetch_b8
        __builtin_prefetch(cols + e + 64, 0, 0);
    }
    long long r = rows[e];
    long long c = cols[e];
    if (r == c) return;  // removed self loops get weight 0
    float norm = dinv[(int)r] * dinv[(int)c];
    float4 v = reinterpret_cast<const float4*>(xw + (size_t)c * DDIM)[lane];
    float* dst = out + (size_t)r * DDIM + lane * 4;
    unsafeAtomicAdd(dst + 0, norm * v.x);  // global_atomic_add_f32
    unsafeAtomicAdd(dst + 1, norm * v.y);
    unsafeAtomicAdd(dst + 2, norm * v.z);
    unsafeAtomicAdd(dst + 3, norm * v.w);
}

// ----------------------------------------------------------------- driver ---
extern "C" void kernel_launch(void* const* d_in, const int* in_sizes, int n_in,
                              void* d_out, int out_size, void* d_ws, size_t ws_size,
                              hipStream_t stream) {
    const float*     x    = (const float*)d_in[0];
    const long long* ei   = (const long long*)d_in[1];  // int64 [2, E]
    const float*     w    = (const float*)d_in[2];
    const float*     bias = (const float*)d_in[3];
    float*           out  = (float*)d_out;

    const int n_nodes = in_sizes[0] / DDIM;
    const int n_edges = in_sizes[1] / 2;
    const long long* rows = ei;
    const long long* cols = ei + n_edges;

    // Workspace layout: xw [N*128 f32] | dinv [N f32] | deg [N i32]
    float* xw   = (float*)d_ws;
    float* dinv = xw + (size_t)n_nodes * DDIM;
    int*   deg  = (int*)(dinv + n_nodes);

    int nb_nodes = (n_nodes + 255) / 256;
    int nb_edges = (n_edges + 255) / 256;

    deg_init_kernel<<<nb_nodes, 256, 0, stream>>>(deg, n_nodes);
    deg_accum_kernel<<<nb_edges, 256, 0, stream>>>(rows, cols, deg, n_edges);
    dinv_kernel<<<nb_nodes, 256, 0, stream>>>(deg, dinv, n_nodes);

    gemm_xw_kernel<<<(n_nodes + 127) / 128, 256, 0, stream>>>(x, w, xw, n_nodes);

    int t_init = n_nodes * 32;  // one thread per float4 chunk
    out_init_kernel<<<(t_init + 255) / 256, 256, 0, stream>>>(xw, dinv, bias, out, n_nodes);

    spmm_edge_kernel<<<(n_edges + 7) / 8, 256, 0, stream>>>(rows, cols, dinv, xw, out, n_edges);
}